// SSDIntNBitTableBatchedEmbeddingBags_21509196219245
// MI455X (gfx1250) — compile-verified
//
#include <hip/hip_runtime.h>

typedef unsigned int       u32;
typedef unsigned long long u64;
typedef u32 v4u  __attribute__((ext_vector_type(4)));
typedef int v8i  __attribute__((ext_vector_type(8)));
typedef int v4i  __attribute__((ext_vector_type(4)));
typedef _Float16 h2 __attribute__((ext_vector_type(2)));

namespace {
constexpr int kT     = 4;        // tables
constexpr int kE     = 1000000;  // rows per table
constexpr int kD     = 64;       // dim
constexpr int kB     = 8192;     // bags per table
constexpr int kWaves = 8;        // bags (waves) per block
constexpr int kChunk = 8;        // rows per TDM gather descriptor (32-bit index mode)
constexpr int kBuf   = 4;        // LDS buffers per wave (pipeline depth 4)
}

// Generic pointer into LDS aperture: low 32 bits are the LDS byte address.
__device__ __forceinline__ u32 lds_offset(const void* p) { return (u32)(u64)p; }

// Issue one TDM gather: fetch `nvalid` rows (64 x int32 each) of the table at
// `table_base` into LDS at `lds_byte`, rows selected by ii[0..7].
__device__ __forceinline__ void tdm_gather_rows(u64 table_base, u32 lds_byte,
                                                int nvalid, const int* ii) {
  // ---- D# group 0 (128b): count=1 | gather_index_size=32b | gather_mode=1,
  //      lds_addr, global_addr[56:0], type=2 ("image")
  v4u g0;
  g0.x = 0xC0000001u;
  g0.y = lds_byte;
  g0.z = (u32)table_base;
  g0.w = ((u32)(table_base >> 32) & 0x01FFFFFFu) | (2u << 30);

  // ---- D# group 1 (256b)
  v8i g1;
  g1[0] = (int)(2u << 16);                            // data_size = 4 bytes
  g1[1] = (int)((u32)kD << 16);                       // tensor_dim0 = 64 (bits 79:48 lo)
  g1[2] = (int)(((u32)kE & 0xFFFFu) << 16);           // tensor_dim1 = E (bits 111:80 lo)
  g1[3] = (int)(((u32)kE >> 16) | ((u32)kD << 16));   // tensor_dim1 hi | tile_dim0 = 64
  g1[4] = nvalid;                                     // tile_dim1 = #valid gather indices
  g1[5] = kD;                                         // tensor_dim0_stride = 64 elements
  g1[6] = 0;
  g1[7] = 0;

  // ---- D# groups 2/3: 32-bit row indices 0..7
  v4i g2 = { ii[0], ii[1], ii[2], ii[3] };
  v4i g3 = { ii[4], ii[5], ii[6], ii[7] };
  // Trailing 8-dword group (6-arg builtin form); mirror the indices.
  v8i g4 = { ii[0], ii[1], ii[2], ii[3], ii[4], ii[5], ii[6], ii[7] };

  __builtin_amdgcn_tensor_load_to_lds(g0, g1, g2, g3, g4, 0);
}

__global__ __launch_bounds__(kWaves * 32, 1) void tbe_tdm_kernel(
    const int* __restrict__ qweights,   // [T,E,D] int32 (uint8 payload)
    const float* __restrict__ scales,   // [T,E]
    const float* __restrict__ biases,   // [T,E]
    const int* __restrict__ indices,    // [T*B*L]
    const int* __restrict__ offsets,    // [T*B+1]
    _Float16* __restrict__ out,         // [B, T*D]
    int n_bags) {
  // Quad-buffered row staging: per wave 4 x 8 rows x 256 B = 8 KB (64 KB/block)
  __shared__ int smem[kWaves * kBuf * kChunk * kD];

  const int lane = threadIdx.x & 31;
  const int wid  = threadIdx.x >> 5;
  const int bag  = __builtin_amdgcn_readfirstlane((int)(blockIdx.x * kWaves + wid));
  if (bag >= n_bags) return;

  const int t  = bag / kB;            // table/feature
  const int bi = bag - t * kB;        // bag within table
  const int start = __builtin_amdgcn_readfirstlane(offsets[bag]);
  const int n     = __builtin_amdgcn_readfirstlane(offsets[bag + 1]) - start;
  const int nchunk = (n + kChunk - 1) / kChunk;

  int* buf0 = &smem[wid * kBuf * kChunk * kD];
  const u32 lds0  = (u32)__builtin_amdgcn_readfirstlane((int)lds_offset(buf0));
  const u64 tbase = (u64)qweights + (u64)t * kE * kD * sizeof(int);

  // Issue gather for chunk `c` into LDS buffer c%4.
  auto issue = [&](int c) {
    const int p   = start + c * kChunk;
    const int rem = n - c * kChunk;
    int ii[kChunk];
#pragma unroll
    for (int j = 0; j < kChunk; ++j)
      ii[j] = (j < rem) ? __builtin_amdgcn_readfirstlane(indices[p + j]) : 0;
    tdm_gather_rows(tbase, lds0 + (u32)((c & (kBuf - 1)) * kChunk * kD * 4),
                    (rem < kChunk ? rem : kChunk), ii);
  };

  // Prime the pipeline: up to 3 gathers in flight before processing starts.
  const int pre = (nchunk < kBuf - 1) ? nchunk : (kBuf - 1);
  for (int c = 0; c < pre; ++c) issue(c);

  float acc0 = 0.f, acc1 = 0.f, bsum = 0.f;
  const float* srow = scales + (u64)t * kE;
  const float* brow = biases + (u64)t * kE;

  for (int c = 0; c < nchunk; ++c) {
    if (c + kBuf - 1 < nchunk) {
      // Target buffer was last read at iteration c-1; make sure those LDS
      // reads retired before the DMA may overwrite it.
      asm volatile("s_wait_dscnt 0" ::: "memory");
      issue(c + kBuf - 1);
      __builtin_amdgcn_s_wait_tensorcnt(3);   // 4 in flight -> oldest (c) done
    } else if (c + 3 == nchunk) {
      __builtin_amdgcn_s_wait_tensorcnt(2);   // drain: 3 outstanding max
    } else if (c + 2 == nchunk) {
      __builtin_amdgcn_s_wait_tensorcnt(1);
    } else {
      __builtin_amdgcn_s_wait_tensorcnt(0);   // last chunk complete
    }

    const int m    = (n - c * kChunk < kChunk) ? (n - c * kChunk) : kChunk;
    const int* buf = buf0 + (c & (kBuf - 1)) * kChunk * kD;
    const int base = start + c * kChunk;

    if (m == kChunk) {
      // Batched scalar metadata: 1 KM wait for indices, 1 for scales/biases
      // (instead of one per row).
      int ix[kChunk];
#pragma unroll
      for (int j = 0; j < kChunk; ++j)
        ix[j] = __builtin_amdgcn_readfirstlane(indices[base + j]);
      float ss[kChunk], bb[kChunk];
#pragma unroll
      for (int j = 0; j < kChunk; ++j) {
        ss[j] = srow[ix[j]];                  // wave-uniform -> scalar load
        bb[j] = brow[ix[j]];
      }
#pragma unroll
      for (int r = 0; r < kChunk; ++r) {
        const int2 q = *(const int2*)(buf + r * kD + lane * 2);  // ds_load_b64
        acc0 = fmaf((float)q.x, ss[r], acc0);
        acc1 = fmaf((float)q.y, ss[r], acc1);
        bsum += bb[r];                        // bias is dim-invariant: add once/row
      }
    } else {
      for (int r = 0; r < m; ++r) {
        const int idx = __builtin_amdgcn_readfirstlane(indices[base + r]);
        const float s = srow[idx];
        const float b = brow[idx];
        const int2 q  = *(const int2*)(buf + r * kD + lane * 2);
        acc0 = fmaf((float)q.x, s, acc0);
        acc1 = fmaf((float)q.y, s, acc1);
        bsum += b;
      }
    }
  }

  h2 hv;
  hv.x = (_Float16)(acc0 + bsum);
  hv.y = (_Float16)(acc1 + bsum);
  _Float16* op = out + (u64)bi * (kT * kD) + (u64)t * kD + lane * 2;
  *(h2*)op = hv;                              // 4-byte coalesced half2 store
}

extern "C" void kernel_launch(void* const* d_in, const int* in_sizes, int n_in,
                              void* d_out, int out_size, void* d_ws, size_t ws_size,
                              hipStream_t stream) {
  const int*   qweights = (const int*)d_in[0];
  const float* scales   = (const float*)d_in[1];
  const float* biases   = (const float*)d_in[2];
  const int*   indices  = (const int*)d_in[3];
  const int*   offsets  = (const int*)d_in[4];
  _Float16*    out      = (_Float16*)d_out;

  const int n_bags = in_sizes[4] - 1;   // T*B = 32768
  const int blocks = (n_bags + kWaves - 1) / kWaves;
  hipLaunchKernelGGL(tbe_tdm_kernel, dim3(blocks), dim3(kWaves * 32), 0, stream,
                     qweights, scales, biases, indices, offsets, out, n_bags);
}